// CombinedLoss_57011395887749
// MI455X (gfx1250) — compile-verified
//
#include <hip/hip_runtime.h>
#include <stdint.h>

typedef float v2f __attribute__((ext_vector_type(2)));
typedef float v8f __attribute__((ext_vector_type(8)));

#define PAD 2
#define TILE 16
#define IMG 1024
#define TPP 4096                 // 64*64 tiles per 1024x1024 plane
#define NPLANES 48               // B*C = 16*3
#define NTILES (NPLANES * TPP)   // 196608
#define WAVES_PER_BLOCK 8
#define NBUCKETS 1024

// 5-tap Gaussian, sigma=1.5, normalized (matches reference _gaussian_kernel).
// Branchless band lookup: returns g[d] for d in [0,4], else 0.
__device__ __forceinline__ float gauss_tap(int d) {
    float r = 0.0f;
    r = (d == 0 || d == 4) ? 0.1200787f : r;
    r = (d == 1 || d == 3) ? 0.2338803f : r;
    r = (d == 2)           ? 0.2920819f : r;
    return r;
}

template<int F>
__device__ __forceinline__ float fval(const float* Ps, const float* Ts, int r, int c) {
    const float p = Ps[r * 20 + c];
    const float t = Ts[r * 20 + c];
    if (F == 0) return p;
    if (F == 1) return t;
    if (F == 2) return p * p;
    if (F == 3) return t * t;
    return p * t;
}

// Separable 5x5 Gaussian blur of one field over a 16x16 tile (20x20 halo in LDS),
// both passes as V_WMMA_F32_16X16X4_F32 band-matrix multiplies. The band fragment
// (Gh column / Gv row) is identical for both passes and hoisted by the caller.
template<int F>
__device__ __forceinline__ v8f blur_field(const float* Ps, const float* Ts, float* Hs,
                                          const v2f* band, int half, int lm) {
    // Horizontal: H(20x16) = In(20x20) x Gh(20x16), as two 16-row groups.
    v8f htop = {};
    v8f hbot = {};
#pragma unroll
    for (int k = 0; k < 5; ++k) {
        const int j0 = 4 * k + 2 * half;      // A K-pair index per half-wave
        v2f aT, aB;
        aT.x = fval<F>(Ps, Ts, lm,     j0);
        aT.y = fval<F>(Ps, Ts, lm,     j0 + 1);
        aB.x = fval<F>(Ps, Ts, lm + 4, j0);
        aB.y = fval<F>(Ps, Ts, lm + 4, j0 + 1);
        htop = __builtin_amdgcn_wmma_f32_16x16x4_f32(false, aT, false, band[k], (short)0, htop, false, false);
        hbot = __builtin_amdgcn_wmma_f32_16x16x4_f32(false, aB, false, band[k], (short)0, hbot, false, false);
    }
    // D layout -> LDS rows: row = v + 8*half, col = lm. htop covers In rows 0..15,
    // hbot covers In rows 4..19 (only its rows 16..19 are new).
#pragma unroll
    for (int v = 0; v < 8; ++v)
        Hs[(v + 8 * half) * 16 + lm] = htop[v];
    if (half) {
#pragma unroll
        for (int v = 4; v < 8; ++v)
            Hs[(v + 12) * 16 + lm] = hbot[v];
    }
    // Vertical: Out(16x16) = Gv(16x20) x H(20x16). A is the same band fragment.
    v8f acc = {};
#pragma unroll
    for (int k = 0; k < 5; ++k) {
        const int i0 = 4 * k + 2 * half;
        v2f b;
        b.x = Hs[i0 * 16 + lm];
        b.y = Hs[(i0 + 1) * 16 + lm];
        acc = __builtin_amdgcn_wmma_f32_16x16x4_f32(false, band[k], false, b, (short)0, acc, false, false);
    }
    return acc;
}

__global__ __launch_bounds__(256) void
ssim_l1_kernel(const float* __restrict__ pred, const float* __restrict__ targ,
               float* __restrict__ bucketL1, float* __restrict__ bucketSSIM) {
    __shared__ float smem[WAVES_PER_BLOCK * 1120];   // per wave: 20x20 P, 20x20 T, 20x16 H
    const int lane = threadIdx.x & 31;
    const int wave = threadIdx.x >> 5;
    const int half = lane >> 4;
    const int lm   = lane & 15;

    float* Ps = smem + wave * 1120;
    float* Ts = Ps + 400;
    float* Hs = Ts + 400;

    const int tile  = blockIdx.x * WAVES_PER_BLOCK + wave;
    const int plane = tile >> 12;
    const int tIdx  = tile & 4095;
    const int row0  = (tIdx >> 6) << 4;
    const int col0  = (tIdx & 63) << 4;

    const float* pp = pred + (size_t)plane * (IMG * IMG);
    const float* tp = targ + (size_t)plane * (IMG * IMG);

    // Gaussian band fragment, shared by horizontal-B and vertical-A (identical
    // per (lane,k) since both are evaluated at (4k + 2*half, lm)).
    v2f band[5];
#pragma unroll
    for (int k = 0; k < 5; ++k) {
        const int j0 = 4 * k + 2 * half;
        band[k].x = gauss_tap(j0     - lm);
        band[k].y = gauss_tap(j0 + 1 - lm);
    }

    // 20x20 halo load. Interior tiles (94%) need no zero padding: use the CDNA5
    // async memory->LDS path (no VGPR staging, tracked by ASYNCcnt). Border
    // tiles use the predicated sync path with zero fill (matches SAME padding).
    const bool interior = (row0 >= PAD) && (row0 + TILE + PAD <= IMG) &&
                          (col0 >= PAD) && (col0 + TILE + PAD <= IMG);
    if (interior) {
        // 20 rows x 10 b64 chunks per array; 8B aligned (col0-2 is even).
        for (int i2 = lane; i2 < 200; i2 += 32) {
            const int r = i2 / 10;
            const int c = (i2 - r * 10) * 2;
            const size_t off = (size_t)(row0 - PAD + r) * IMG + (col0 - PAD + c);
            const uint32_t ldsP = (uint32_t)(uintptr_t)(Ps + r * 20 + c);
            const uint32_t ldsT = (uint32_t)(uintptr_t)(Ts + r * 20 + c);
            const float* gp = pp + off;
            const float* gt = tp + off;
            asm volatile("global_load_async_to_lds_b64 %0, %1, off"
                         :: "v"(ldsP), "v"(gp) : "memory");
            asm volatile("global_load_async_to_lds_b64 %0, %1, off"
                         :: "v"(ldsT), "v"(gt) : "memory");
        }
        asm volatile("s_wait_asynccnt 0x0" ::: "memory");
    } else {
        for (int idx = lane; idx < 400; idx += 32) {
            const int r  = idx / 20;
            const int c  = idx - r * 20;
            const int gr = row0 - PAD + r;
            const int gc = col0 - PAD + c;
            const bool ok = ((unsigned)gr < (unsigned)IMG) && ((unsigned)gc < (unsigned)IMG);
            float pv = 0.0f, tv = 0.0f;
            if (ok) {
                const size_t off = (size_t)gr * IMG + gc;
                pv = pp[off];
                tv = tp[off];
            }
            Ps[idx] = pv;
            Ts[idx] = tv;
        }
    }
    // Same-wave LDS ops are in-order (DScnt); no cross-wave sharing -> no barrier.

    const v8f b_p  = blur_field<0>(Ps, Ts, Hs, band, half, lm);   // mu_p
    const v8f b_t  = blur_field<1>(Ps, Ts, Hs, band, half, lm);   // mu_t
    const v8f b_pp = blur_field<2>(Ps, Ts, Hs, band, half, lm);   // E[p^2]
    const v8f b_tt = blur_field<3>(Ps, Ts, Hs, band, half, lm);   // E[t^2]
    const v8f b_pt = blur_field<4>(Ps, Ts, Hs, band, half, lm);   // E[p*t]

    float ssimSum = 0.0f, l1Sum = 0.0f;
#pragma unroll
    for (int v = 0; v < 8; ++v) {
        const float mp = b_p[v], mt = b_t[v];
        const float mp2 = mp * mp, mt2 = mt * mt, mpt = mp * mt;
        const float sp  = b_pp[v] - mp2;
        const float st  = b_tt[v] - mt2;
        const float spt = b_pt[v] - mpt;
        const float num = (2.0f * mpt + 1.0e-4f) * (2.0f * spt + 9.0e-4f);
        const float den = (mp2 + mt2 + 1.0e-4f) * (sp + st + 9.0e-4f);
        ssimSum += num / den;
        // L1 over the central 16x16 (same D-layout mapping).
        const int r = v + 8 * half + PAD;
        const int c = lm + PAD;
        l1Sum += fabsf(Ps[r * 20 + c] - Ts[r * 20 + c]);
    }

    // wave32 butterfly reduction
#pragma unroll
    for (int off = 16; off > 0; off >>= 1) {
        ssimSum += __shfl_xor(ssimSum, off, 32);
        l1Sum   += __shfl_xor(l1Sum,   off, 32);
    }
    if (lane == 0) {
        const int bucket = tile & (NBUCKETS - 1);
        atomicAdd(&bucketL1[bucket],   l1Sum);
        atomicAdd(&bucketSSIM[bucket], ssimSum);
    }
}

__global__ void init_ws_kernel(float* ws) {
    const int i = blockIdx.x * blockDim.x + threadIdx.x;
    if (i < 2 * NBUCKETS) ws[i] = 0.0f;
}

__global__ void finalize_kernel(const float* __restrict__ bucketL1,
                                const float* __restrict__ bucketSSIM,
                                float* __restrict__ out) {
    __shared__ float sA[256], sB[256];
    float a = 0.0f, b = 0.0f;
    for (int i = threadIdx.x; i < NBUCKETS; i += 256) {
        a += bucketL1[i];
        b += bucketSSIM[i];
    }
    sA[threadIdx.x] = a;
    sB[threadIdx.x] = b;
    __syncthreads();
    for (int s = 128; s > 0; s >>= 1) {
        if ((int)threadIdx.x < s) {
            sA[threadIdx.x] += sA[threadIdx.x + s];
            sB[threadIdx.x] += sB[threadIdx.x + s];
        }
        __syncthreads();
    }
    if (threadIdx.x == 0) {
        const float invN = 1.0f / 50331648.0f;   // 16*3*1024*1024
        const float l1   = sA[0] * invN;
        const float ssim = sB[0] * invN;
        out[0] = 0.9f * l1 + 0.1f * (1.0f - ssim);
    }
}

extern "C" void kernel_launch(void* const* d_in, const int* in_sizes, int n_in,
                              void* d_out, int out_size, void* d_ws, size_t ws_size,
                              hipStream_t stream) {
    (void)in_sizes; (void)n_in; (void)out_size; (void)ws_size;
    const float* pred = (const float*)d_in[0];
    const float* targ = (const float*)d_in[1];
    float* ws         = (float*)d_ws;
    float* bucketL1   = ws;
    float* bucketSSIM = ws + NBUCKETS;

    init_ws_kernel<<<(2 * NBUCKETS + 255) / 256, 256, 0, stream>>>(ws);
    ssim_l1_kernel<<<NTILES / WAVES_PER_BLOCK, 256, 0, stream>>>(pred, targ, bucketL1, bucketSSIM);
    finalize_kernel<<<1, 256, 0, stream>>>(bucketL1, bucketSSIM, (float*)d_out);
}